// Model_82325933130446
// MI455X (gfx1250) — compile-verified
//
#include <hip/hip_runtime.h>
#include <math.h>

#define NU 50000
#define NI 50000
#define NT 100000
#define NE 2000000

typedef __attribute__((ext_vector_type(2))) float v2f;
typedef __attribute__((ext_vector_type(8))) float v8f;

// ---------- order-preserving float<->uint mapping for atomic max ----------
__device__ __forceinline__ unsigned f2ord(float f) {
  unsigned u = __float_as_uint(f);
  return (u & 0x80000000u) ? ~u : (u | 0x80000000u);
}
__device__ __forceinline__ float ord2f(unsigned u) {
  return (u & 0x80000000u) ? __uint_as_float(u ^ 0x80000000u)
                           : __uint_as_float(~u);
}

__device__ __forceinline__ float wave_sum(float x) {
  #pragma unroll
  for (int o = 16; o > 0; o >>= 1) x += __shfl_xor(x, o, 32);
  return x;
}
__device__ __forceinline__ float wave_max(float x) {
  #pragma unroll
  for (int o = 16; o > 0; o >>= 1) x = fmaxf(x, __shfl_xor(x, o, 32));
  return x;
}

// ---------------------------- zero scratch ----------------------------
__global__ void k_zero(float* p, int n) {
  int i = blockIdx.x * blockDim.x + threadIdx.x;
  if (i < n) p[i] = 0.0f;
}

// ------------------- f_src0 = relu(uv @ W_src_0 + b) -------------------
__global__ void k_fsrc0(const float* __restrict__ uv, const float* __restrict__ W,
                        const float* __restrict__ b, float* __restrict__ fsrc0) {
  __shared__ float suv[256];
  int t = threadIdx.x;                 // 128 threads
  suv[t] = uv[t];
  suv[t + 128] = uv[t + 128];
  __syncthreads();
  float s = 0.0f;
  #pragma unroll 8
  for (int k = 0; k < 256; ++k) s += suv[k] * W[k * 128 + t];
  fsrc0[t] = fmaxf(s + b[t], 0.0f);
}

// ------------- s_item[i] = dot(uv, emb_item[i]) / sqrt(256) -------------
__global__ void k_sitem(const float* __restrict__ uv, const float* __restrict__ emb,
                        float* __restrict__ s_item) {
  __shared__ float suv[256];
  int t = threadIdx.x;                 // 256 threads = 8 waves
  suv[t] = uv[t];
  __syncthreads();
  int wid = t >> 5, lane = t & 31;
  int item = blockIdx.x * 8 + wid;     // NI/8 = 6250 blocks exactly
  const float* row = emb + (size_t)(NU + item) * 256 + lane * 8;
  float4 e0 = *(const float4*)(row);
  float4 e1 = *(const float4*)(row + 4);
  int k0 = lane * 8;
  float p = e0.x * suv[k0] + e0.y * suv[k0 + 1] + e0.z * suv[k0 + 2] + e0.w * suv[k0 + 3]
          + e1.x * suv[k0 + 4] + e1.y * suv[k0 + 5] + e1.z * suv[k0 + 6] + e1.w * suv[k0 + 7];
  p = wave_sum(p);
  if (lane == 0) s_item[item] = p * (1.0f / 16.0f);
}

// ------------- C = relu(A[MxK] @ B[KxN] + bias), fp32 WMMA -------------
// One wave computes one 16x16 tile via V_WMMA_F32_16X16X4_F32 over K/4 steps.
// M % 16 == 0, N % 16 == 0 guaranteed by caller (50000 = 3125*16).
__global__ void k_gemm_wmma(const float* __restrict__ A, const float* __restrict__ B,
                            const float* __restrict__ bias, float* __restrict__ C,
                            int K, int N) {
  int wid  = threadIdx.x >> 5;               // 4 waves / block
  int lane = threadIdx.x & 31;
  int l    = lane & 15;
  int half = lane >> 4;
  int ntile = blockIdx.y * 4 + wid;
  if (ntile * 16 >= N) return;               // wave-uniform
  int m0 = blockIdx.x * 16;
  int n0 = ntile * 16;

  v8f acc = {};
  const float* Arow = A + (size_t)(m0 + l) * K;
  for (int k = 0; k < K; k += 4) {
    int ka = k + 2 * half;
    v2f a, b;
    a.x = Arow[ka];
    a.y = Arow[ka + 1];
    b.x = B[(size_t)ka * N + n0 + l];
    b.y = B[(size_t)(ka + 1) * N + n0 + l];
    acc = __builtin_amdgcn_wmma_f32_16x16x4_f32(false, a, false, b,
                                                (short)0, acc, false, false);
  }
  float bs = bias[n0 + l];
  #pragma unroll
  for (int r = 0; r < 8; ++r) {
    int row = m0 + r + 8 * half;
    C[(size_t)row * N + n0 + l] = fmaxf(acc[r] + bs, 0.0f);
  }
}

// ---------------- layer-0 softmax: max over s_item[dst[e]] ----------------
__global__ void k_max0(const int* __restrict__ edst, const float* __restrict__ s_item,
                       unsigned* __restrict__ maxp) {
  int e = blockIdx.x * blockDim.x + threadIdx.x;
  float a = (e < NE) ? s_item[edst[e]] : -3.4e38f;
  a = wave_max(a);
  if ((threadIdx.x & 31) == 0) atomicMax(maxp, f2ord(a));
}

__global__ void k_sum0(const int* __restrict__ edst, const float* __restrict__ s_item,
                       const float* __restrict__ scal, float* __restrict__ sump) {
  float m = ord2f(*(const unsigned*)&scal[0]);
  int e = blockIdx.x * blockDim.x + threadIdx.x;
  float s = (e < NE) ? __expf(s_item[edst[e]] - m) : 0.0f;
  s = wave_sum(s);
  if ((threadIdx.x & 31) == 0) atomicAdd(sump, s);
}

// --- layer-0 scatter: h1_user[src] += f_dst0[dst]*w ; w_item0[dst] += w ---
__global__ void k_scatter0(const int* __restrict__ esrc, const int* __restrict__ edst,
                           const float* __restrict__ s_item, const float* __restrict__ fdst0,
                           const float* __restrict__ scal, float* __restrict__ h1,
                           float* __restrict__ w_item0) {
  int wid = threadIdx.x >> 5, lane = threadIdx.x & 31;
  int e = blockIdx.x * 8 + wid;               // NE/8 = 250000 blocks exactly
  int src = esrc[e], dst = edst[e];
  float m = ord2f(*(const unsigned*)&scal[0]);
  float w = __expf(s_item[dst] - m) * (1.0f / scal[1]);
  if (lane == 0) atomicAdd(&w_item0[dst], w);
  const float* fr = fdst0 + (size_t)dst * 128 + lane * 4;
  __builtin_prefetch(fdst0 + (size_t)dst * 128 + ((lane * 4 + 64) & 127), 0, 1);
  float4 v = *(const float4*)fr;
  float* ur = h1 + (size_t)src * 128 + lane * 4;
  atomicAdd(ur + 0, v.x * w);
  atomicAdd(ur + 1, v.y * w);
  atomicAdd(ur + 2, v.z * w);
  atomicAdd(ur + 3, v.w * w);
}

// ---------- h1_item[i] = w_item0[i] * f_src0  (dense write) ----------
__global__ void k_item0(const float* __restrict__ w_item0, const float* __restrict__ fsrc0,
                        float* __restrict__ h1) {
  int idx = blockIdx.x * blockDim.x + threadIdx.x;   // NI*128 threads
  int i = idx >> 7, c = idx & 127;
  h1[(size_t)(NU + i) * 128 + c] = w_item0[i] * fsrc0[c];
}

// ------- alpha1[e] = dot(h1u[src], h1i[dst]) / sqrt(128); track max -------
__global__ void k_alpha1(const int* __restrict__ esrc, const int* __restrict__ edst,
                         const float* __restrict__ h1, float* __restrict__ alpha1,
                         unsigned* __restrict__ maxp) {
  __shared__ float smax[8];
  int wid = threadIdx.x >> 5, lane = threadIdx.x & 31;
  int e = blockIdx.x * 8 + wid;
  int src = esrc[e], dst = edst[e];
  const float* ar = h1 + (size_t)src * 128 + lane * 4;
  const float* br = h1 + (size_t)(NU + dst) * 128 + lane * 4;
  float4 a = *(const float4*)ar;
  float4 b = *(const float4*)br;
  float p = a.x * b.x + a.y * b.y + a.z * b.z + a.w * b.w;
  p = wave_sum(p) * 0.08838834764831845f;            // 1/sqrt(128)
  if (lane == 0) { alpha1[e] = p; smax[wid] = p; }
  __syncthreads();
  if (threadIdx.x == 0) {
    float m = smax[0];
    #pragma unroll
    for (int i = 1; i < 8; ++i) m = fmaxf(m, smax[i]);
    atomicMax(maxp, f2ord(m));
  }
}

__global__ void k_sum1(const float* __restrict__ alpha1, const float* __restrict__ scal,
                       float* __restrict__ sump) {
  float m = ord2f(*(const unsigned*)&scal[2]);
  int e = blockIdx.x * blockDim.x + threadIdx.x;
  float s = (e < NE) ? __expf(alpha1[e] - m) : 0.0f;
  s = wave_sum(s);
  if ((threadIdx.x & 31) == 0) atomicAdd(sump, s);
}

// - layer-1 scatter: h2u[src] += f_dst1[dst]*w ; h2i[dst] += f_src1[src]*w -
__global__ void k_scatter1(const int* __restrict__ esrc, const int* __restrict__ edst,
                           const float* __restrict__ alpha1, const float* __restrict__ fsrc1,
                           const float* __restrict__ fdst1, const float* __restrict__ scal,
                           float* __restrict__ h2) {
  int wid = threadIdx.x >> 5, lane = threadIdx.x & 31;
  int e = blockIdx.x * 8 + wid;
  int src = esrc[e], dst = edst[e];
  float m = ord2f(*(const unsigned*)&scal[2]);
  float w = __expf(alpha1[e] - m) * (1.0f / scal[3]);
  const float* fd = fdst1 + (size_t)dst * 64 + lane * 2;
  const float* fs = fsrc1 + (size_t)src * 64 + lane * 2;
  __builtin_prefetch(fd, 0, 1);
  float2 v = *(const float2*)fd;
  float2 g = *(const float2*)fs;
  float* ur = h2 + (size_t)src * 64 + lane * 2;
  float* ir = h2 + (size_t)(NU + dst) * 64 + lane * 2;
  atomicAdd(ur + 0, v.x * w);
  atomicAdd(ur + 1, v.y * w);
  atomicAdd(ir + 0, g.x * w);
  atomicAdd(ir + 1, g.y * w);
}

// --- out[t] = [emb[t](256) | l2norm(h1[t])(128) | l2norm(h2[t])(64)] ---
__global__ void k_final(const float* __restrict__ emb, const float* __restrict__ h1,
                        const float* __restrict__ h2, float* __restrict__ out) {
  int wid = threadIdx.x >> 5, lane = threadIdx.x & 31;
  int t = blockIdx.x * 8 + wid;               // NT/8 = 12500 blocks
  float* o = out + (size_t)t * 448;
  const float* er = emb + (size_t)t * 256;
  *(float4*)(o + lane * 4)       = *(const float4*)(er + lane * 4);
  *(float4*)(o + 128 + lane * 4) = *(const float4*)(er + 128 + lane * 4);
  float4 v = *(const float4*)(h1 + (size_t)t * 128 + lane * 4);
  float ss = wave_sum(v.x * v.x + v.y * v.y + v.z * v.z + v.w * v.w);
  float inv = 1.0f / fmaxf(sqrtf(ss), 1e-12f);
  float4 vn = { v.x * inv, v.y * inv, v.z * inv, v.w * inv };
  *(float4*)(o + 256 + lane * 4) = vn;
  float2 u = *(const float2*)(h2 + (size_t)t * 64 + lane * 2);
  float ss2 = wave_sum(u.x * u.x + u.y * u.y);
  float inv2 = 1.0f / fmaxf(sqrtf(ss2), 1e-12f);
  float2 un = { u.x * inv2, u.y * inv2 };
  *(float2*)(o + 384 + lane * 2) = un;
}

// ------------------------------- driver -------------------------------
extern "C" void kernel_launch(void* const* d_in, const int* in_sizes, int n_in,
                              void* d_out, int out_size, void* d_ws, size_t ws_size,
                              hipStream_t stream) {
  const float* emb      = (const float*)d_in[0];
  const float* uv       = (const float*)d_in[1];
  const float* W_src_0  = (const float*)d_in[2];
  const float* b_src_0  = (const float*)d_in[3];
  const float* W_dst_0  = (const float*)d_in[4];
  const float* b_dst_0  = (const float*)d_in[5];
  const float* W_src_1  = (const float*)d_in[6];
  const float* b_src_1  = (const float*)d_in[7];
  const float* W_dst_1  = (const float*)d_in[8];
  const float* b_dst_1  = (const float*)d_in[9];
  const int*   esrc     = (const int*)d_in[10];
  const int*   edst     = (const int*)d_in[11];
  float* out = (float*)d_out;

  float* ws = (float*)d_ws;
  float* h1      = ws;                         // NT*128
  float* h2      = h1 + (size_t)NT * 128;      // NT*64
  float* w_item0 = h2 + (size_t)NT * 64;       // NI
  float* scal    = w_item0 + NI;               // [max0(ord), sum0, max1(ord), sum1]
  const int ZCOUNT = NT * 128 + NT * 64 + NI + 4;
  float* fdst0  = scal + 4;                    // NI*128
  float* fsrc1  = fdst0 + (size_t)NI * 128;    // NU*64
  float* fdst1  = fsrc1 + (size_t)NU * 64;     // NI*64
  float* s_item = fdst1 + (size_t)NI * 64;     // NI
  float* alpha1 = s_item + NI;                 // NE
  float* fsrc0  = alpha1 + NE;                 // 128

  // 1. zero accumulators + softmax scalars (0u lies below ord(-inf))
  k_zero<<<(ZCOUNT + 255) / 256, 256, 0, stream>>>(ws, ZCOUNT);
  // 2. tiny GEMV f_src0
  k_fsrc0<<<1, 128, 0, stream>>>(uv, W_src_0, b_src_0, fsrc0);
  // 3. per-item scores
  k_sitem<<<NI / 8, 256, 0, stream>>>(uv, emb, s_item);
  // 4. f_dst0 = relu(emb_items @ W_dst_0 + b)   [50000x256x128] WMMA
  k_gemm_wmma<<<dim3(NI / 16, 2), 128, 0, stream>>>(emb + (size_t)NU * 256, W_dst_0,
                                                    b_dst_0, fdst0, 256, 128);
  // 5-6. global softmax over edges (layer 0)
  k_max0<<<(NE + 255) / 256, 256, 0, stream>>>(edst, s_item, (unsigned*)&scal[0]);
  k_sum0<<<(NE + 255) / 256, 256, 0, stream>>>(edst, s_item, scal, &scal[1]);
  // 7. user-side scatter + item weight accumulation
  k_scatter0<<<NE / 8, 256, 0, stream>>>(esrc, edst, s_item, fdst0, scal, h1, w_item0);
  // 8. item rows = outer(w_item0, f_src0)
  k_item0<<<NI * 128 / 256, 256, 0, stream>>>(w_item0, fsrc0, h1);
  // 9. layer-1 feature GEMMs [50000x128x64] WMMA
  k_gemm_wmma<<<dim3(NU / 16, 1), 128, 0, stream>>>(h1, W_src_1, b_src_1, fsrc1, 128, 64);
  k_gemm_wmma<<<dim3(NI / 16, 1), 128, 0, stream>>>(h1 + (size_t)NU * 128, W_dst_1,
                                                    b_dst_1, fdst1, 128, 64);
  // 10-11. edge dots + global softmax (layer 1)
  k_alpha1<<<NE / 8, 256, 0, stream>>>(esrc, edst, h1, alpha1, (unsigned*)&scal[2]);
  k_sum1<<<(NE + 255) / 256, 256, 0, stream>>>(alpha1, scal, &scal[3]);
  // 12. dual scatter into h2
  k_scatter1<<<NE / 8, 256, 0, stream>>>(esrc, edst, alpha1, fsrc1, fdst1, scal, h2);
  // 13. fused copy + L2-normalize epilogue
  k_final<<<NT / 8, 256, 0, stream>>>(emb, h1, h2, out);
}